// GraphDistillationNetwork_43559558316733
// MI455X (gfx1250) — compile-verified
//
#include <hip/hip_runtime.h>
#include <hip/hip_bf16.h>
#include <math.h>

#define N_NODES 50000
#define N_EDGES 800000
#define HIDDEN  128
#define NUM_GDL 2
#define NCOLS   (3 * HIDDEN)   // 384 fused output columns

typedef __attribute__((ext_vector_type(2))) float v2f;
typedef __attribute__((ext_vector_type(8))) float v8f;

// ---------------------------------------------------------------- utilities

__global__ void fill_zero_kernel(float* __restrict__ p, size_t n) {
    size_t t = (size_t)blockIdx.x * blockDim.x + threadIdx.x;
    if (t < n) p[t] = 0.0f;
}

// deg[v] = # edges with col == v  (float, matches reference segment_sum of ones)
__global__ void deg_kernel(const int* __restrict__ col, float* __restrict__ degf) {
    int e = blockIdx.x * blockDim.x + threadIdx.x;
    if (e < N_EDGES) atomicAdd(&degf[col[e]], 1.0f);
}

__global__ void invdeg_kernel(const float* __restrict__ degf, float* __restrict__ invdeg) {
    int v = blockIdx.x * blockDim.x + threadIdx.x;
    if (v < N_NODES) invdeg[v] = 1.0f / degf[v];
}

// s[i] = sum over incoming edges of 1/deg[row_e]
__global__ void s_kernel(const int* __restrict__ row, const int* __restrict__ col,
                         const float* __restrict__ invdeg, float* __restrict__ sbuf) {
    int e = blockIdx.x * blockDim.x + threadIdx.x;
    if (e < N_EDGES) atomicAdd(&sbuf[col[e]], invdeg[row[e]]);
}

// Pack fused weight, PAIR-MAJOR so a B fragment is one 8-byte load:
//   logical w(k,c):  c<128 : W1[c][k] ; c<256 : W2[c-128][k] ; else W2[c-256][128+k]
//   stored at WcP[(k>>1)*768 + c*2 + (k&1)]  -> (w(2t,c), w(2t+1,c)) adjacent.
__global__ void pack_w_kernel(const float* __restrict__ W1, const float* __restrict__ W2,
                              float* __restrict__ WcP) {
    int t = blockIdx.x * blockDim.x + threadIdx.x;
    if (t >= HIDDEN * NCOLS) return;
    int k = t / NCOLS, c = t % NCOLS;
    float v;
    if (c < HIDDEN)            v = W1[c * HIDDEN + k];
    else if (c < 2 * HIDDEN)   v = W2[(c - HIDDEN) * (2 * HIDDEN) + k];
    else                       v = W2[(c - 2 * HIDDEN) * (2 * HIDDEN) + HIDDEN + k];
    WcP[(k >> 1) * (2 * NCOLS) + c * 2 + (k & 1)] = v;
}

// ---------------------------------------------------------------- fused GEMM
// Y = X[nrows,128] @ W[128,384]; columns split into xw1 / pi / pjq buffers.
// pjq store is pre-scaled by invdeg[row] (folds per-edge norm into the source).
// Block = 64 threads (2 waves), 64 rows. Each wave owns TWO 16-row M-tiles so
// every b64 B-fragment load feeds two V_WMMA_F32_16X16X4_F32 (0.5 loads/wmma,
// and halves the L2 weight traffic).
#define LDS_STRIDE 132  // 128 + 4 pad: b64 frag reads hit all 64 banks once

__global__ __launch_bounds__(64)
void gemm_fused_kernel(const float* __restrict__ X, const float* __restrict__ WcP,
                       const float* __restrict__ invdeg,
                       float* __restrict__ xw1, float* __restrict__ pi,
                       float* __restrict__ pjq, int nrows) {
    __shared__ float tileA[64][LDS_STRIDE];

    const int tid  = threadIdx.x;
    const int wave = tid >> 5;
    const int lane = tid & 31;
    const int base = blockIdx.x * 64;

    // Stage 64x128 A tile with float4 loads (rows are 512B -> 16B aligned).
    #pragma unroll
    for (int i = 0; i < 32; ++i) {
        int idx = i * 64 + tid;        // 0..2047 float4 slots
        int r   = idx >> 5;
        int c4  = idx & 31;
        float4 v = {0.f, 0.f, 0.f, 0.f};
        int g = base + r;
        if (g < nrows) v = reinterpret_cast<const float4*>(X + (size_t)g * HIDDEN)[c4];
        reinterpret_cast<float4*>(&tileA[r][0])[c4] = v;
    }
    __syncthreads();

    // A fragments for two 16-row tiles across the whole K=128 panel.
    // Lane layout (ISA 7.12.2): lanes 0-15 -> M=lane, K={4kk,4kk+1}; lanes 16-31 -> K={4kk+2,4kk+3}.
    const int m0    = wave * 32 + (lane & 15);
    const int m1    = m0 + 16;
    const int khalf = (lane >> 4) * 2;
    v2f a0[32], a1[32];
    #pragma unroll
    for (int kk = 0; kk < 32; ++kk) {
        int k0 = kk * 4 + khalf;
        a0[kk] = *reinterpret_cast<const v2f*>(&tileA[m0][k0]);
        a1[kk] = *reinterpret_cast<const v2f*>(&tileA[m1][k0]);
    }

    const int ncl = lane & 15;
    for (int nt = 0; nt < 24; ++nt) {
        v8f acc0 = {}, acc1 = {};
        const float* wp = WcP + (size_t)(nt * 16 + ncl) * 2;
        #pragma unroll
        for (int kk = 0; kk < 32; ++kk) {
            int k2 = kk * 2 + (lane >> 4);            // pair row: K = {2k2, 2k2+1}
            v2f b = *reinterpret_cast<const v2f*>(wp + (size_t)k2 * (2 * NCOLS));
            acc0 = __builtin_amdgcn_wmma_f32_16x16x4_f32(
                false, a0[kk], false, b, (short)0, acc0, false, false);
            acc1 = __builtin_amdgcn_wmma_f32_16x16x4_f32(
                false, a1[kk], false, b, (short)0, acc1, false, false);
        }
        // D layout: VGPR v, lanes 0-15 -> M=v, lanes 16-31 -> M=8+v; N=lane&15.
        float* dst; int c;
        if (nt < 8)       { dst = xw1; c = nt * 16 + ncl; }
        else if (nt < 16) { dst = pi;  c = (nt - 8) * 16 + ncl; }
        else              { dst = pjq; c = (nt - 16) * 16 + ncl; }
        const int mBase = base + wave * 32 + (lane >> 4) * 8;
        #pragma unroll
        for (int v = 0; v < 8; ++v) {
            int g0 = mBase + v;
            int g1 = g0 + 16;
            if (g0 < nrows) {
                float val = acc0[v];
                if (nt >= 16) val *= invdeg[g0];
                dst[(size_t)g0 * HIDDEN + c] = val;
            }
            if (g1 < nrows) {
                float val = acc1[v];
                if (nt >= 16) val *= invdeg[g1];
                dst[(size_t)g1 * HIDDEN + c] = val;
            }
        }
    }
}

// ---------------------------------------------------------------- edge scatter
// agg2[col_e] += pjq[row_e]; one wave per edge, float4 gather + 4 f32 atomics/lane.
__global__ __launch_bounds__(256)
void scatter_kernel(const int* __restrict__ row, const int* __restrict__ col,
                    const float* __restrict__ pjq, float* __restrict__ agg2) {
    const int wave = threadIdx.x >> 5;
    const int lane = threadIdx.x & 31;
    const int e = blockIdx.x * 8 + wave;
    if (e >= N_EDGES) return;
    const int r = row[e];
    const int c = col[e];
    const float4 q = reinterpret_cast<const float4*>(pjq + (size_t)r * HIDDEN)[lane];
    float* dst = agg2 + (size_t)c * HIDDEN + lane * 4;
    atomicAdd(dst + 0, q.x);
    atomicAdd(dst + 1, q.y);
    atomicAdd(dst + 2, q.z);
    atomicAdd(dst + 3, q.w);
}

// ---------------------------------------------------------------- finalize
// x' = gelu(xw1 - s[i]*pi - agg2 + b)   (exact erf gelu)
__global__ void finalize_kernel(const float* __restrict__ xw1, const float* __restrict__ pi,
                                const float* __restrict__ agg2, const float* __restrict__ sbuf,
                                const float* __restrict__ bias, float* __restrict__ xout) {
    size_t t = (size_t)blockIdx.x * blockDim.x + threadIdx.x;
    if (t >= (size_t)N_NODES * HIDDEN) return;
    int i = (int)(t / HIDDEN);
    int c = (int)(t % HIDDEN);
    float v = xw1[t] - sbuf[i] * pi[t] - agg2[t] + bias[c];
    xout[t] = 0.5f * v * (1.0f + erff(v * 0.7071067811865476f));
}

// ---------------------------------------------------------------- launch

extern "C" void kernel_launch(void* const* d_in, const int* in_sizes, int n_in,
                              void* d_out, int out_size, void* d_ws, size_t ws_size,
                              hipStream_t stream) {
    const float* x_in = (const float*)d_in[0];
    const int*   ei   = (const int*)d_in[1];
    const float* W1s  = (const float*)d_in[2];
    const float* W2s  = (const float*)d_in[3];
    const float* bs   = (const float*)d_in[4];

    const int* row = ei;             // sources
    const int* col = ei + N_EDGES;   // targets

    const size_t n  = N_NODES;
    const size_t nd = n * HIDDEN;
    float* ws     = (float*)d_ws;
    float* degf   = ws;              // n
    float* invdeg = degf + n;        // n
    float* sbuf   = invdeg + n;      // n
    float* xw1    = sbuf + n;        // n*128
    float* pi     = xw1 + nd;        // n*128
    float* pjq    = pi + nd;         // n*128
    float* agg2   = pjq + nd;        // n*128
    float* xmid   = agg2 + nd;       // n*128
    float* WcP    = xmid + nd;       // 128*384 pair-packed

    const int T = 256;
    auto blocks = [](size_t work, int t) { return (int)((work + t - 1) / t); };

    // Graph-only precompute (degrees, per-target norm sum).
    fill_zero_kernel<<<blocks(n, T), T, 0, stream>>>(degf, n);
    fill_zero_kernel<<<blocks(n, T), T, 0, stream>>>(sbuf, n);
    deg_kernel<<<blocks(N_EDGES, T), T, 0, stream>>>(col, degf);
    invdeg_kernel<<<blocks(n, T), T, 0, stream>>>(degf, invdeg);
    s_kernel<<<blocks(N_EDGES, T), T, 0, stream>>>(row, col, invdeg, sbuf);

    const float* xcur = x_in;
    for (int k = 0; k < NUM_GDL; ++k) {
        pack_w_kernel<<<blocks(HIDDEN * NCOLS, T), T, 0, stream>>>(
            W1s + (size_t)k * HIDDEN * HIDDEN, W2s + (size_t)k * HIDDEN * 2 * HIDDEN, WcP);
        fill_zero_kernel<<<blocks(nd, T), T, 0, stream>>>(agg2, nd);

        gemm_fused_kernel<<<(N_NODES + 63) / 64, 64, 0, stream>>>(
            xcur, WcP, invdeg, xw1, pi, pjq, N_NODES);

        scatter_kernel<<<(N_EDGES + 7) / 8, 256, 0, stream>>>(row, col, pjq, agg2);

        float* xo = (k == NUM_GDL - 1) ? (float*)d_out : xmid;
        finalize_kernel<<<blocks(nd, T), T, 0, stream>>>(
            xw1, pi, agg2, sbuf, bs + (size_t)k * HIDDEN, xo);
        xcur = xo;
    }
}